// SparseConv_24610162606296
// MI455X (gfx1250) — compile-verified
//
#include <hip/hip_runtime.h>
#include <math.h>

typedef __attribute__((ext_vector_type(2))) float v2f;
typedef __attribute__((ext_vector_type(8))) float v8f;

#define NPTS   (4 * 32768)
#define CIN    64
#define COUT   64
#define GRID_H 256
#define GRID_W 256
#define NBATCH 4
#define MT     4   // M-tiles (16 points each) per wave
#define NT     4   // N-tiles (16 out-ch each) per wave -> 64x64 per wave

// ---------------- workspace layout ----------------
// [0    .. 8   )  : unsigned int minbits[2]
// [128  .. 384 )  : float zrow[64]        (all-zero A row for invalid taps)
// [1024 .. +1MB)  : float2 sxy[NPTS]
// [..   .. +1MB)  : int2   cell[NPTS]
// [..   .. +1MB)  : int    grid[4*256*256]
// [..   .. +144KB): float  wtr[9][64][64]  (weights transposed to [tap][cout][cin])

__global__ void k_init(unsigned int* __restrict__ minbits,
                       float* __restrict__ zrow,
                       int* __restrict__ grid) {
    int i = blockIdx.x * blockDim.x + threadIdx.x;
    if (i < 2) minbits[i] = 0x7f800000u;  // +inf bit pattern
    if (i < CIN) zrow[i] = 0.0f;
    if (i < NBATCH * GRID_H * GRID_W) grid[i] = -1;
}

// wtr[tap][n][k] = weight[tap][k][n]  -> B fragment becomes one contiguous b64
__global__ void k_wtranspose(const float* __restrict__ w, float* __restrict__ wtr) {
    int i = blockIdx.x * blockDim.x + threadIdx.x;  // over 9*64*64
    int tap = i >> 12;
    int k   = (i >> 6) & 63;
    int n   = i & 63;
    wtr[tap * 4096 + n * 64 + k] = w[i];
}

__global__ void k_sigmoid_min(const float* __restrict__ anchor,
                              float2* __restrict__ sxy,
                              unsigned int* __restrict__ minbits) {
    __shared__ unsigned int smin[2];
    if (threadIdx.x == 0) { smin[0] = 0x7f800000u; smin[1] = 0x7f800000u; }
    __syncthreads();

    int p = blockIdx.x * blockDim.x + threadIdx.x;
    float a0 = anchor[p * 2 + 0];
    float a1 = anchor[p * 2 + 1];
    a0 = fminf(fmaxf(a0, -9.21f), 9.21f);
    a1 = fminf(fmaxf(a1, -9.21f), 9.21f);
    float s0 = 1.0f / (1.0f + __expf(-a0));
    float s1 = 1.0f / (1.0f + __expf(-a1));
    sxy[p] = make_float2(s0, s1);

    atomicMin(&smin[0], __float_as_uint(s0));  // positive floats: uint order == float order
    atomicMin(&smin[1], __float_as_uint(s1));
    __syncthreads();
    if (threadIdx.x == 0) {
        atomicMin(&minbits[0], smin[0]);
        atomicMin(&minbits[1], smin[1]);
    }
}

__global__ void k_scatter(const float2* __restrict__ sxy,
                          const unsigned int* __restrict__ minbits,
                          int2* __restrict__ cell,
                          int* __restrict__ grid) {
    int p = blockIdx.x * blockDim.x + threadIdx.x;
    float mn0 = __uint_as_float(minbits[0]);
    float mn1 = __uint_as_float(minbits[1]);
    float2 s = sxy[p];
    int iy = (int)((s.x - mn0) * 256.0f);
    int ix = (int)((s.y - mn1) * 256.0f);
    iy = min(iy, GRID_H - 1);
    ix = min(ix, GRID_W - 1);
    cell[p] = make_int2(iy, ix);
    int b = p >> 15;  // 32768 points per batch
    atomicMax(&grid[b * (GRID_H * GRID_W) + iy * GRID_W + ix], p);
}

// One wave computes a 64x64 output macro-tile (4 M-tiles x 4 N-tiles, 16 acc
// tiles). Per k-step: 4 A-fragments + 4 B-fragments (each one b64 load) feed
// 16 WMMAs -> 0.5 loads per WMMA. Branchless gathers (clamped grid coords,
// zero-row fallback). Block = 8 waves = 512 points.
__global__ void __launch_bounds__(256) k_conv_wmma(
    const float* __restrict__ feats,   // [NPTS][64]
    const float* __restrict__ wtr,     // [9][64][64] transposed (tap, cout, cin)
    const int2* __restrict__ cell,     // [NPTS] (iy, ix)
    const int* __restrict__ grid,      // [4][256][256]
    const float* __restrict__ zrow,    // [64] zeros
    float* __restrict__ out)           // [NPTS][64]
{
    int lane    = threadIdx.x & 31;
    int wave    = threadIdx.x >> 5;
    int pt_base = (blockIdx.x * 8 + wave) * 64;  // this wave's 64 points
    int l15     = lane & 15;
    int khalf   = lane >> 4;  // 0 -> K pair {4s,4s+1}; 1 -> {4s+2,4s+3}

    // Per-M-tile point state carried by this lane (A row = pt_base+16t+l15)
    int2 c[MT];
    const int* gb[MT];
#pragma unroll
    for (int t = 0; t < MT; ++t) {
        int p = pt_base + 16 * t + l15;
        c[t]  = cell[p];
        gb[t] = grid + (p >> 15) * (GRID_H * GRID_W);
    }

    // Per-N-tile B row base for this lane: wtr[tap][j*16+l15][*]
    // (tap offset added in the loop)
    v8f acc[MT][NT];
#pragma unroll
    for (int t = 0; t < MT; ++t)
#pragma unroll
        for (int j = 0; j < NT; ++j) acc[t][j] = (v8f){};

    for (int tap = 0; tap < 9; ++tap) {
        int dy = tap / 3 - 1, dx = tap % 3 - 1;
        const float* wtb = wtr + tap * CIN * COUT;  // [n][k]

        const float* arow[MT];
#pragma unroll
        for (int t = 0; t < MT; ++t) {
            int ny = c[t].x + dy, nx = c[t].y + dx;
            bool inb = (ny >= 0) && (ny < GRID_H) && (nx >= 0) && (nx < GRID_W);
            int nyc = min(max(ny, 0), GRID_H - 1);
            int nxc = min(max(nx, 0), GRID_W - 1);
            int nid = gb[t][nyc * GRID_W + nxc];          // unconditional load
            bool valid = inb && (nid >= 0);
            arow[t] = valid ? (feats + (long)max(nid, 0) * CIN) : zrow;
        }

#pragma unroll
        for (int s = 0; s < 16; ++s) {
            int k = 4 * s + khalf * 2;                    // even -> 8B aligned
            v2f a[MT], bv[NT];
#pragma unroll
            for (int t = 0; t < MT; ++t)
                a[t] = *(const v2f*)(arow[t] + k);        // one b64 gather
#pragma unroll
            for (int j = 0; j < NT; ++j)
                bv[j] = *(const v2f*)(wtb + (j * 16 + l15) * CIN + k);  // one b64
#pragma unroll
            for (int t = 0; t < MT; ++t)
#pragma unroll
                for (int j = 0; j < NT; ++j)
                    acc[t][j] = __builtin_amdgcn_wmma_f32_16x16x4_f32(
                        false, a[t], false, bv[j], (short)0, acc[t][j],
                        false, false);
        }
    }

    // C/D layout: lanes 0-15 -> M=0..7 in v0..v7, lanes 16-31 -> M=8..15
    int mrow0 = khalf * 8;
#pragma unroll
    for (int t = 0; t < MT; ++t) {
#pragma unroll
        for (int r = 0; r < 8; ++r) {
            long rbase = (long)(pt_base + 16 * t + mrow0 + r) * COUT + l15;
#pragma unroll
            for (int j = 0; j < NT; ++j)
                out[rbase + j * 16] = acc[t][j][r];
        }
    }
}

extern "C" void kernel_launch(void* const* d_in, const int* in_sizes, int n_in,
                              void* d_out, int out_size, void* d_ws, size_t ws_size,
                              hipStream_t stream) {
    const float* feats  = (const float*)d_in[0];  // instance_feature [4,32768,64]
    const float* anchor = (const float*)d_in[1];  // anchor           [4,32768,2]
    const float* weight = (const float*)d_in[2];  // weight           [3,3,64,64]
    float* out = (float*)d_out;

    char* ws = (char*)d_ws;
    unsigned int* minbits = (unsigned int*)ws;
    float*  zrow = (float*)(ws + 128);
    float2* sxy  = (float2*)(ws + 1024);
    int2*   cell = (int2*)(ws + 1024 + (size_t)NPTS * 8);
    int*    grid = (int*)(ws + 1024 + (size_t)NPTS * 16);
    float*  wtr  = (float*)(ws + 1024 + (size_t)NPTS * 16
                                 + (size_t)NBATCH * GRID_H * GRID_W * 4);

    (void)in_sizes; (void)n_in; (void)out_size; (void)ws_size;

    k_init<<<(NBATCH * GRID_H * GRID_W + 255) / 256, 256, 0, stream>>>(minbits, zrow, grid);
    k_wtranspose<<<(9 * CIN * COUT + 255) / 256, 256, 0, stream>>>(weight, wtr);
    k_sigmoid_min<<<NPTS / 256, 256, 0, stream>>>(anchor, sxy, minbits);
    k_scatter<<<NPTS / 256, 256, 0, stream>>>(sxy, minbits, cell, grid);
    k_conv_wmma<<<NPTS / 512, 256, 0, stream>>>(feats, wtr, cell, grid, zrow, out);
}